// C_idht_52132313039072
// MI455X (gfx1250) — compile-verified
//
#include <hip/hip_runtime.h>
#include <math.h>

#ifndef M_PI
#define M_PI 3.14159265358979323846
#endif

typedef unsigned int u32x4 __attribute__((ext_vector_type(4)));
typedef int          i32x4 __attribute__((ext_vector_type(4)));
typedef int          i32x8 __attribute__((ext_vector_type(8)));
typedef float        f32x2 __attribute__((ext_vector_type(2)));
typedef float        f32x4 __attribute__((ext_vector_type(4)));

// Problem constants (setup_inputs: N=4, C=128, A=180, R=184, H=W=128)
#define NCH   512             // N*C flattened channels
#define A_    180
#define R_    184
#define H_    128
#define W_    128
// Tiling
#define G_    8               // channels per workgroup
#define AB_   6               // angles per LDS block (180 = 30 * 6)
#define NBLK  (A_ / AB_)      // 30
#define TPB_  256
#define PPT_  8               // pixels per thread (same w, h step = 2)
#define TILE_PIX (TPB_ * PPT_)        // 2048
#define ST_   ((H_ * W_) / TILE_PIX)  // 8 spatial tiles
#define BLK_ELEMS (AB_ * R_)          // 1104 (a,r) cells per block
#define BUF_ELEMS (G_ * BLK_ELEMS)    // 8832 floats = 35328 B
// LDS layout (floats): cos/sin pairs | raw (TDM dest) | fmt x2 (interleaved)
#define CS_OFF   0                       // 180 float2 = 360 floats
#define RAW_OFF  384
#define FMT_OFF  (RAW_OFF + BUF_ELEMS)   // 9216
#define SMEM_FLOATS (FMT_OFF + 2 * BUF_ELEMS)  // 26880 floats = 107520 B

#if defined(__has_builtin)
#if __has_builtin(__builtin_amdgcn_tensor_load_to_lds)
#define HAS_TDM 1
#endif
#endif

#ifdef HAS_TDM
// One 3D TDM load per block: tile = [G_ channels][AB_ angle rows][R_ rho] f32.
// Global strides: rho contiguous, angle-row stride R_, channel stride A_*R_.
// LDS destination compact [G_][AB_][R_].
__device__ __forceinline__ void tdm_issue(const float* gsrc, unsigned ldsOff) {
  unsigned long long ga = (unsigned long long)(const void*)gsrc;
  u32x4 g0 = {
    1u,                                               // count = 1
    ldsOff,                                           // LDS byte address
    (unsigned)(ga & 0xffffffffu),                     // global_addr[31:0]
    (unsigned)((ga >> 32) & 0x1ffffffu) | (2u << 30)  // global_addr[56:32] | type=2
  };
  i32x8 g1 = {
    (int)(2u << 16),                  // mask=0, data_size=2 (4B)
    (int)((unsigned)R_ << 16),        // tensor_dim0[15:0]=184
    (int)((unsigned)AB_ << 16),       // tensor_dim0 hi=0 ; tensor_dim1[15:0]=6
    (int)((unsigned)R_ << 16),        // tensor_dim1 hi=0 ; tile_dim0=184
    (int)(((unsigned)G_ << 16) | (unsigned)AB_),  // tile_dim1=6 ; tile_dim2=8
    (int)R_,                          // tensor_dim0_stride = 184
    (int)(((unsigned)(A_ * R_) & 0xffffu) << 16), // d1_stride[15:0]=33120
    (int)((unsigned)(A_ * R_) >> 16)  // d1_stride hi
  };
  i32x4 g2 = { G_, 0, 0, 0 };         // tensor_dim2 = 8
  i32x4 g3 = { 0, 0, 0, 0 };
#if __clang_major__ >= 23
  i32x8 g4 = { 0, 0, 0, 0, 0, 0, 0, 0 };
  __builtin_amdgcn_tensor_load_to_lds(g0, g1, g2, g3, g4, 0);
#else
  __builtin_amdgcn_tensor_load_to_lds(g0, g1, g2, g3, 0);
#endif
}
#endif

__global__ __launch_bounds__(TPB_)
void idht_kernel(const float* __restrict__ acc, float* __restrict__ out) {
  extern __shared__ float smem[];
  f32x2* csT  = (f32x2*)(smem + CS_OFF);   // {cos, sin} per angle
  float* raw  = smem + RAW_OFF;            // TDM destination [G][AB][R]
  float* fmt  = smem + FMT_OFF;            // interleaved [AB][R][G], x2 buffers

  const int tid = threadIdx.x;
  const int st  = blockIdx.x;
  const int ch0 = blockIdx.y * G_;

  if (tid < A_) {
    double sv, cv;
    sincos((double)tid * (M_PI / 180.0), &sv, &cv);
    f32x2 cs = { (float)cv, (float)sv };
    csT[tid] = cs;
  }

  // Thread's PPT_ pixels: p_i = pix0 + i*256 -> same w, h steps by 2.
  const int pix0 = st * TILE_PIX + tid;
  const float xc  = (float)((pix0 & (W_ - 1)) - (W_ / 2));
  const float yc0 = (float)((pix0 >> 7) - (H_ / 2));

  // Packed accumulators: accv2[i][j] = channels {2j, 2j+1} for pixel i
  f32x2 accv2[PPT_][G_ / 2];
#pragma unroll
  for (int i = 0; i < PPT_; ++i)
#pragma unroll
    for (int j = 0; j < G_ / 2; ++j) accv2[i][j] = (f32x2){0.0f, 0.0f};

  const float* gbase = acc + (size_t)ch0 * (A_ * R_);

#ifdef HAS_TDM
  if (tid < 32) {
    tdm_issue(gbase, (unsigned)(unsigned long long)(void*)raw);
  }
#endif

  for (int k = 0; k < NBLK; ++k) {
    float* fmtk = fmt + (size_t)(k & 1) * BUF_ELEMS;

#ifdef HAS_TDM
    if (tid < 32) {
      __builtin_amdgcn_s_wait_tensorcnt(0);   // block k landed in raw
    }
    __syncthreads();  // raw ready; prior readers of fmt[k&1] done (barrier chain)
#else
    __syncthreads();
    for (int e = tid; e < BUF_ELEMS; e += TPB_) {
      int g   = e / BLK_ELEMS;
      int rem = e - g * BLK_ELEMS;
      raw[e] = gbase[(size_t)g * (A_ * R_) + (size_t)k * BLK_ELEMS + rem];
    }
    __syncthreads();
#endif

    // Re-layout raw [g][e] -> fmt [e][g]  (e = aa*184 + r), b128 stores.
    for (int u = tid; u < 2 * BLK_ELEMS; u += TPB_) {
      int e    = u >> 1;
      int half = u & 1;            // channels half*4 .. half*4+3
      const float* rb = raw + half * 4 * BLK_ELEMS + e;
      f32x4 v = { rb[0 * BLK_ELEMS], rb[1 * BLK_ELEMS],
                  rb[2 * BLK_ELEMS], rb[3 * BLK_ELEMS] };
      *(f32x4*)(fmtk + (size_t)e * G_ + half * 4) = v;
    }
    __syncthreads();  // fmt[k&1] ready; raw free for next DMA

#ifdef HAS_TDM
    if (tid < 32 && (k + 1) < NBLK) {
      // Next block DMA overlaps this block's compute.
      tdm_issue(gbase + (size_t)(k + 1) * BLK_ELEMS,
                (unsigned)(unsigned long long)(void*)raw);
    }
#endif

#pragma unroll
    for (int aa = 0; aa < AB_; ++aa) {
      int a = k * AB_ + aa;
      f32x2 cs = csT[a];                    // one ds_load_b64
      const float* rowb = fmtk + (size_t)aa * (R_ * G_);
      // t0 for pixel i=0; pixel i adds (2i)*sin via one fma (exact constant)
      float t0 = fmaf(cs.x, xc, fmaf(cs.y, yc0, (float)(R_ / 2)));
#pragma unroll
      for (int i = 0; i < PPT_; ++i) {
        float t = (i == 0) ? t0 : fmaf(cs.y, (float)(2 * i), t0);
        int r = (int)rintf(t);              // v_rndne + v_cvt (round-half-even)
        unsigned ur = (unsigned)r;
        float m  = (ur < (unsigned)R_) ? 1.0f : 0.0f;
        int   rc = (ur < (unsigned)R_) ? r : 0;
        f32x2 m2 = { m, m };
        const f32x4* src = (const f32x4*)(rowb + (size_t)rc * G_);
        f32x4 v0 = src[0];                  // ds_load_b128: channels 0..3
        f32x4 v1 = src[1];                  // ds_load_b128: channels 4..7
        accv2[i][0] = __builtin_elementwise_fma(m2, v0.lo, accv2[i][0]);  // v_pk_fma_f32
        accv2[i][1] = __builtin_elementwise_fma(m2, v0.hi, accv2[i][1]);
        accv2[i][2] = __builtin_elementwise_fma(m2, v1.lo, accv2[i][2]);
        accv2[i][3] = __builtin_elementwise_fma(m2, v1.hi, accv2[i][3]);
      }
    }
  }

  // Coalesced output stores: lanes cover consecutive w
#pragma unroll
  for (int j = 0; j < G_ / 2; ++j) {
    float* og0 = out + (size_t)(ch0 + 2 * j + 0) * (H_ * W_) + pix0;
    float* og1 = out + (size_t)(ch0 + 2 * j + 1) * (H_ * W_) + pix0;
#pragma unroll
    for (int i = 0; i < PPT_; ++i) {
      og0[i * TPB_] = accv2[i][j].x;
      og1[i * TPB_] = accv2[i][j].y;
    }
  }
}

extern "C" void kernel_launch(void* const* d_in, const int* in_sizes, int n_in,
                              void* d_out, int out_size, void* d_ws, size_t ws_size,
                              hipStream_t stream) {
  (void)in_sizes; (void)n_in; (void)d_ws; (void)ws_size; (void)out_size;
  const float* acc = (const float*)d_in[0];
  float* out = (float*)d_out;

  // 107520 B dynamic LDS (> default 64KB cap; CDNA5 WGP has 320KB -> 3 WGs/WGP)
  (void)hipFuncSetAttribute((const void*)idht_kernel,
                            hipFuncAttributeMaxDynamicSharedMemorySize,
                            SMEM_FLOATS * 4);

  dim3 grid(ST_, NCH / G_, 1);   // 8 spatial tiles x 64 channel groups = 512 WGs
  dim3 block(TPB_, 1, 1);        // 8 waves (wave32)
  idht_kernel<<<grid, block, SMEM_FLOATS * 4, stream>>>(acc, out);
}